// RepeatWordPositionalEncoding_9947144257848
// MI455X (gfx1250) — compile-verified
//
#include <hip/hip_runtime.h>
#include <stdint.h>

// RepeatWordPositionalEncoding for MI455X (gfx1250).
// out[t,b,c] = x[t,b,c] + (t < total_dur[b] ? pe[wid(b,t), c] : 0)
// Memory-bound: ~1 GB HBM streaming -> ~44us floor at 23.3 TB/s.
// pe (10 MB) stays L2-resident; x/out use non-temporal hints so the stream
// does not evict it. WMMA is not applicable without precision loss; the
// CDNA5-specific path used here is async global->LDS + s_wait_asynccnt.

typedef float v4f __attribute__((ext_vector_type(4)));

#define T_DIM 4096
#define B_DIM 64
#define C_DIM 512
#define W_DIM 256
#define C4_DIM (C_DIM / 4) // 128 float4 per row

// ---------------------------------------------------------------------------
// Kernel 1: per-batch duration cumsum + word-id table.
// One block per batch. Durations staged into LDS with the CDNA5 async
// global->LDS path (GVS mode: saddr base + 32-bit vgpr offset), then an
// inclusive scan and a binary search per timestep.
// ---------------------------------------------------------------------------
__global__ __launch_bounds__(W_DIM) void wid_kernel(const int* __restrict__ dur,
                                                    short* __restrict__ wid) {
  __shared__ int scum[W_DIM];
  const int b = blockIdx.x;
  const int j = threadIdx.x;

  // Async copy dur[b, j] -> scum[j]. Each lane moves 4 bytes directly into
  // LDS; issuing wave waits on its own ASYNCcnt, barrier makes it visible
  // workgroup-wide. Generic __shared__ pointer low 32 bits == LDS offset.
  {
    unsigned lds_dst = (unsigned)(uintptr_t)(&scum[j]);
    unsigned voff    = (unsigned)((b * W_DIM + j) * 4u);
    asm volatile("global_load_async_to_lds_b32 %0, %1, %2\n\t"
                 "s_wait_asynccnt 0x0"
                 :: "v"(lds_dst), "v"(voff), "s"(dur)
                 : "memory");
  }
  __syncthreads();

  // Inclusive scan (Hillis-Steele) over the 256 durations.
  for (int off = 1; off < W_DIM; off <<= 1) {
    int t = (j >= off) ? scum[j - off] : 0;
    __syncthreads();
    scum[j] += t;
    __syncthreads();
  }
  const int total = scum[W_DIM - 1];

  // word_id(t) = searchsorted(cum, t, 'right') = first idx with cum[idx] > t.
  for (int t = j; t < T_DIM; t += W_DIM) {
    short w = -1;
    if (t < total) {
      int lo = 0, hi = W_DIM;
      while (lo < hi) {
        int mid = (lo + hi) >> 1;
        if (scum[mid] <= t) lo = mid + 1; else hi = mid;
      }
      w = (short)lo; // <= 255 when t < total
    }
    wid[b * T_DIM + t] = w;
  }
}

// ---------------------------------------------------------------------------
// Kernel 2: streaming add. Each block handles 8 consecutive (t,b) rows
// (rows are contiguous 512-float segments of x/out), 256 threads, one
// float4 per thread per iteration -> 16B/lane fully-coalesced b128 ops.
// x/out: non-temporal (streamed once). pe: temporal (L2-resident gather).
// ---------------------------------------------------------------------------
__global__ __launch_bounds__(256) void add_pe_kernel(const v4f* __restrict__ x,
                                                     const float* __restrict__ pe,
                                                     const short* __restrict__ wid,
                                                     v4f* __restrict__ out) {
  const unsigned base = blockIdx.x * 1024u; // 8 rows * 128 float4
#pragma unroll
  for (int i = 0; i < 4; ++i) {
    unsigned e  = base + (unsigned)i * 256u + threadIdx.x;
    unsigned r  = e >> 7;   // row = t*B + b
    unsigned c4 = e & 127u;
    unsigned t  = r >> 6;   // / B_DIM
    unsigned b  = r & 63u;  // % B_DIM
    int w = wid[b * T_DIM + t];

    v4f xv = __builtin_nontemporal_load(&x[e]);
    v4f pv = {0.0f, 0.0f, 0.0f, 0.0f};
    if (w >= 0) {
      const v4f* per = (const v4f*)(pe + (size_t)w * C_DIM);
      pv = per[c4];
    }
    v4f ov = xv + pv;
    __builtin_nontemporal_store(ov, &out[e]);
  }
}

extern "C" void kernel_launch(void* const* d_in, const int* in_sizes, int n_in,
                              void* d_out, int out_size, void* d_ws, size_t ws_size,
                              hipStream_t stream) {
  const float* x   = (const float*)d_in[0];       // [T, B, C] f32
  const float* pe  = (const float*)d_in[1];       // [5000, C] f32
  const int*   dur = (const int*)d_in[2];         // [B, W] i32
  (void)in_sizes; (void)n_in; (void)out_size; (void)ws_size;

  short* wid = (short*)d_ws;                      // [B, T] = 512 KB scratch

  wid_kernel<<<B_DIM, W_DIM, 0, stream>>>(dur, wid);

  const int rows_per_block = 8;
  const int nblocks = (T_DIM * B_DIM) / rows_per_block; // 32768
  add_pe_kernel<<<nblocks, 256, 0, stream>>>((const v4f*)x, pe, wid, (v4f*)d_out);
}